// GINN_inputLayer_63462436766160
// MI455X (gfx1250) — compile-verified
//
#include <hip/hip_runtime.h>
#include <cstdint>

// ---------------------------------------------------------------------------
// out[b,k] = tanh( sum_{m: seg[m]==k} x[b,m]*w[m] ),  seg sorted, B=256,
// M=65536, K=4096.  Memory-bound (~68 MiB @ 23.3 TB/s ~ 3us).  Strategy:
//   * stream x through LDS with CDNA5 async global->LDS loads, double-buffered
//   * per-thread contiguous 16-element window + run-compaction (sorted seg)
//     -> ~2 global f32 atomics per 16 elements instead of 16
//   * separate epilogue kernel applies gfx1250 v_tanh_f32 in place
// ---------------------------------------------------------------------------

#define BS 256
#define EPT 16                 // contiguous elements per thread per chunk
#define CHUNK (BS * EPT)       // 4096 floats per chunk
#define NITER 4                // chunks per block (double-buffered)
#define LDS_STRIDE 20          // dwords per thread: 16 data + 4 pad (16B align kept)

__global__ __launch_bounds__(BS) void seg_zero_kernel(float4* out4, int n4) {
    int i = blockIdx.x * BS + threadIdx.x;
    if (i < n4) out4[i] = make_float4(0.f, 0.f, 0.f, 0.f);
}

__device__ __forceinline__ void stage_chunk_async(const float* gp, unsigned lds_byte) {
    // 4x b128: inst offset applies to BOTH global and LDS addresses (ISA 08 §4.4)
    asm volatile(
        "global_load_async_to_lds_b128 %0, %1, off\n\t"
        "global_load_async_to_lds_b128 %0, %1, off offset:16\n\t"
        "global_load_async_to_lds_b128 %0, %1, off offset:32\n\t"
        "global_load_async_to_lds_b128 %0, %1, off offset:48"
        :: "v"(lds_byte), "v"(gp) : "memory");
}

__global__ __launch_bounds__(BS) void seg_accum_kernel(const float* __restrict__ x,
                                                       const float* __restrict__ w,
                                                       const int*   __restrict__ seg,
                                                       float*       __restrict__ out,
                                                       int M, int K) {
    __shared__ float xbuf[2][BS * LDS_STRIDE];

    const int t = threadIdx.x;
    const int b = blockIdx.y;
    const int mBlock = blockIdx.x * (CHUNK * NITER);

    const float* xrow = x + (size_t)b * (size_t)M;
    float*       orow = out + (size_t)b * (size_t)K;

    // LDS byte offsets for this lane's private staging slots
    const unsigned lds0 = (unsigned)(uintptr_t)&xbuf[0][t * LDS_STRIDE];
    const unsigned lds1 = (unsigned)(uintptr_t)&xbuf[1][t * LDS_STRIDE];

    // stage chunk 0
    stage_chunk_async(xrow + mBlock + t * EPT, lds0);

    #pragma unroll
    for (int it = 0; it < NITER; ++it) {
        const int mBase = mBlock + it * CHUNK;

        // prefetch next chunk into the other buffer (async engine)
        if (it + 1 < NITER) {
            stage_chunk_async(xrow + mBase + CHUNK + t * EPT,
                              ((it + 1) & 1) ? lds1 : lds0);
        }

        // w / seg for this chunk: direct vector loads, L2-resident across the
        // 256 batches that re-read them (192 MB L2 >> 512 KiB)
        const float4* w4p = (const float4*)(w + mBase + t * EPT);
        const int4*   s4p = (const int4*)(seg + mBase + t * EPT);
        float4 w4[4];
        int4   s4[4];
        #pragma unroll
        for (int g = 0; g < 4; ++g) { w4[g] = w4p[g]; s4[g] = s4p[g]; }

        // async loads retire in order per wave: after issuing the prefetch
        // (outstanding = 8), <=4 outstanding means the current chunk landed.
        if (it + 1 < NITER)
            asm volatile("s_wait_asynccnt 4" ::: "memory");
        else
            asm volatile("s_wait_asynccnt 0" ::: "memory");

        const float4* xb = (const float4*)((it & 1) ? &xbuf[1][t * LDS_STRIDE]
                                                    : &xbuf[0][t * LDS_STRIDE]);

        // run-compaction over this thread's 16 sorted-contiguous elements
        int   curSeg = s4[0].x;
        float run    = 0.f;
        #pragma unroll
        for (int g = 0; g < 4; ++g) {
            float4 x4 = xb[g];
            float px[4] = { x4.x * w4[g].x, x4.y * w4[g].y,
                            x4.z * w4[g].z, x4.w * w4[g].w };
            int   ss[4] = { s4[g].x, s4[g].y, s4[g].z, s4[g].w };
            #pragma unroll
            for (int j = 0; j < 4; ++j) {
                if (ss[j] != curSeg) {
                    unsafeAtomicAdd(&orow[curSeg], run);   // global_atomic_add_f32 (no rtn)
                    run    = 0.f;
                    curSeg = ss[j];
                }
                run += px[j];
            }
        }
        unsafeAtomicAdd(&orow[curSeg], run);
    }
}

__global__ __launch_bounds__(BS) void seg_tanh_kernel(float* out, int n) {
    int i = blockIdx.x * BS + threadIdx.x;
    if (i < n) {
        float v = out[i];
        float r;
        asm volatile("v_tanh_f32 %0, %1" : "=v"(r) : "v"(v));  // gfx1250 trans op
        out[i] = r;
    }
}

extern "C" void kernel_launch(void* const* d_in, const int* in_sizes, int n_in,
                              void* d_out, int out_size, void* d_ws, size_t ws_size,
                              hipStream_t stream) {
    const float* x   = (const float*)d_in[0];   // (B,1,M) f32
    const float* w   = (const float*)d_in[1];   // (M,)    f32
    const int*   seg = (const int*)d_in[2];     // (M,)    i32 sorted
    float*       out = (float*)d_out;           // (B,K)   f32

    const int M = in_sizes[1];
    const int B = in_sizes[0] / M;
    const int K = out_size / B;

    // 1) zero accumulators (harness poisons d_out; we re-zero every call)
    const int n4 = out_size / 4;
    seg_zero_kernel<<<(n4 + BS - 1) / BS, BS, 0, stream>>>((float4*)out, n4);

    // 2) multiply + segment-sum via async-staged streaming + run-compacted atomics
    dim3 grid(M / (CHUNK * NITER), B);
    seg_accum_kernel<<<grid, BS, 0, stream>>>(x, w, seg, out, M, K);

    // 3) tanh in place
    seg_tanh_kernel<<<(out_size + BS - 1) / BS, BS, 0, stream>>>(out, out_size);
}